// LGCN_70368744178404
// MI455X (gfx1250) — compile-verified
//
#include <hip/hip_runtime.h>

#define N_NODES 20000
#define RP 8
#define EMB 64
#define N_CLASSES 16
#define NT 250000
#define NRP (N_NODES * RP)            /* 160000 */
#define NE  (NT * RP)                 /* 2,000,000 */

typedef float v2f __attribute__((ext_vector_type(2)));
typedef float v8f __attribute__((ext_vector_type(8)));

// D = A(16x4,f32) * B(4x16,f32) + C(16x16,f32)
// Lane layouts (CDNA5 ISA 7.12.2):
//   A: a[j] @ lane l -> (m = l&15, k = 2*(l>>4)+j)
//   B: b[j] @ lane l -> (k = 2*(l>>4)+j, n = l&15)
//   D: c[v] @ lane l -> (m = v + 8*(l>>4), n = l&15)
__device__ __forceinline__ v8f wmma4(v2f a, v2f b, v8f c) {
    return __builtin_amdgcn_wmma_f32_16x16x4_f32(
        /*neg_a=*/false, a, /*neg_b=*/false, b,
        /*c_mod=*/(short)0, c, /*reuse_a=*/false, /*reuse_b=*/false);
}

// ---------------------------------------------------------------- zero scratch
__global__ void k_zero(float4* __restrict__ p, long n4) {
    long i = (long)blockIdx.x * blockDim.x + threadIdx.x;
    long stride = (long)gridDim.x * blockDim.x;
    float4 z; z.x = z.y = z.z = z.w = 0.f;
    for (; i < n4; i += stride) p[i] = z;
}

// ------------------------------------------------- l1 / l2 via WMMA f32 16x16x4
// Each wave: 16 triples. B packs W_l1 (cols 0..7) and W_l2 (cols 8..15).
// Softmax over RP for l2 is an 8-lane shfl_xor reduction (lanes 8-15 / 24-31).
__global__ __launch_bounds__(256) void k_l1l2(
    const float* __restrict__ rm, const float* __restrict__ Wl1,
    const float* __restrict__ bl1, const float* __restrict__ Wl2,
    const float* __restrict__ bl2, float* __restrict__ l1T,
    float* __restrict__ l2T)
{
    __shared__ float Wc[64 * 16];
    __shared__ float l1s[RP][128];
    __shared__ float l2s[RP][128];
    const int tid = threadIdx.x;
    for (int idx = tid; idx < 1024; idx += 256) {
        int k = idx >> 4, n = idx & 15;
        Wc[idx] = (n < 8) ? Wl1[k * 8 + n] : Wl2[k * 8 + (n - 8)];
    }
    __syncthreads();

    const int wave = tid >> 5, lane = tid & 31;
    const int m = lane & 15, g = lane >> 4;
    const int tile = blockIdx.x * 8 + wave;
    const int numTiles = NT / 16;                  // 15625, exact

    if (tile < numTiles) {                         // wave-uniform guard
        const int t0 = tile * 16;
        const float* arow = rm + (size_t)(t0 + m) * 64;
        v8f c = {0.f, 0.f, 0.f, 0.f, 0.f, 0.f, 0.f, 0.f};
        #pragma unroll
        for (int k0 = 0; k0 < 64; k0 += 4) {
            const int kk = k0 + 2 * g;
            float2 av = *(const float2*)(arow + kk);
            v2f a; a.x = av.x; a.y = av.y;
            v2f b; b.x = Wc[kk * 16 + m]; b.y = Wc[(kk + 1) * 16 + m];
            c = wmma4(a, b, c);
        }
        const float bia = (m < 8) ? bl1[m] : bl2[m - 8];
        #pragma unroll
        for (int v = 0; v < 8; ++v) {
            float x = c[v] + bia;
            // softmax across the 8 l2 lanes of this row (subgroup xor 1,2,4)
            float mx = x;
            mx = fmaxf(mx, __shfl_xor(mx, 1));
            mx = fmaxf(mx, __shfl_xor(mx, 2));
            mx = fmaxf(mx, __shfl_xor(mx, 4));
            float ex = __expf(x - mx);
            float sm = ex;
            sm += __shfl_xor(sm, 1);
            sm += __shfl_xor(sm, 2);
            sm += __shfl_xor(sm, 4);
            float l2v = fmaxf(ex / sm, 0.f);       // relu (no-op for softmax)
            const int tl = wave * 16 + v + 8 * g;  // local triple index
            if (m < 8) l1s[m][tl] = x;
            else       l2s[m - 8][tl] = l2v;
        }
    }
    __syncthreads();
    const int blockT0 = blockIdx.x * 128;
    int validT = NT - blockT0; if (validT > 128) validT = 128;
    for (int idx = tid; idx < 1024; idx += 256) {
        int r = idx >> 7, i = idx & 127;
        if (i < validT) {
            l1T[(size_t)r * NT + blockT0 + i] = l1s[r][i];
            l2T[(size_t)r * NT + blockT0 + i] = l2s[r][i];
        }
    }
}

// ------------------------------------------------------ colsum / rowsum scatter
// Index 0 is a structural hot spot (whole r=0 block) -> wave-local accumulate.
__global__ void k_sums(const float* __restrict__ l1T, const float* __restrict__ l2T,
                       const int* __restrict__ hcol, const int* __restrict__ vrow,
                       float* __restrict__ colsum, float* __restrict__ rowsum)
{
    long i = (long)blockIdx.x * blockDim.x + threadIdx.x;
    long stride = (long)gridDim.x * blockDim.x;
    float lc = 0.f, lr = 0.f;
    for (long e = i; e < NE; e += stride) {
        int hc = hcol[e]; float v1 = l1T[e];
        if (hc == 0) lc += v1; else atomicAdd(colsum + hc, v1);
        int vr = vrow[e]; float v2 = l2T[e];
        if (vr == 0) lr += v2; else atomicAdd(rowsum + vr, v2);
    }
    lc += __shfl_xor(lc, 1);  lr += __shfl_xor(lr, 1);
    lc += __shfl_xor(lc, 2);  lr += __shfl_xor(lr, 2);
    lc += __shfl_xor(lc, 4);  lr += __shfl_xor(lr, 4);
    lc += __shfl_xor(lc, 8);  lr += __shfl_xor(lr, 8);
    lc += __shfl_xor(lc, 16); lr += __shfl_xor(lr, 16);
    if ((threadIdx.x & 31) == 0) {
        atomicAdd(colsum, lc);
        atomicAdd(rowsum, lr);
    }
}

// -------------------------------------------- h scatter (reduce over r first)
// All 8 r-copies of triple t target h[s[t]] -> register-reduce, 1 atomic pass.
__global__ __launch_bounds__(256) void k_hscatter(
    const float* __restrict__ l1T, const int* __restrict__ hrow,
    const int* __restrict__ hcol, const float* __restrict__ colsum,
    const float* __restrict__ w1, float* __restrict__ h)
{
    const int t = (blockIdx.x * 256 + threadIdx.x) >> 5;   // grid exact: NT/8 blocks
    const int lane = threadIdx.x & 31;
    float ax = 0.f, ay = 0.f;
    #pragma unroll
    for (int r = 0; r < RP; ++r) {
        const size_t e = (size_t)r * NT + t;
        const float lv = l1T[e];
        const int hc = hcol[e];
        const float wgt = lv / colsum[hc];
        const float2 wv = *(const float2*)(w1 + (size_t)hc * EMB + 2 * lane);
        ax = fmaf(wgt, wv.x, ax);
        ay = fmaf(wgt, wv.y, ay);
    }
    const int s = hrow[t];
    atomicAdd(h + (size_t)s * EMB + 2 * lane, ax);
    atomicAdd(h + (size_t)s * EMB + 2 * lane + 1, ay);
}

// -------------------------------------------------------------- h = relu(h+b1)
__global__ void k_hrelu(float* __restrict__ h, const float* __restrict__ bias1) {
    const int i = blockIdx.x * blockDim.x + threadIdx.x;
    if (i < N_NODES * EMB) h[i] = fmaxf(h[i] + bias1[i & 63], 0.f);
}

// ------------------------------------------ h2 scatter (row 0 via LDS reducer)
__global__ __launch_bounds__(256) void k_h2scatter(
    const float* __restrict__ l2T, const int* __restrict__ vrow,
    const int* __restrict__ vcol, const float* __restrict__ rowsum,
    const float* __restrict__ h, float* __restrict__ h2)
{
    __shared__ float acc0[EMB];
    const int tid = threadIdx.x;
    if (tid < EMB) acc0[tid] = 0.f;
    __syncthreads();
    const int t = (blockIdx.x * 256 + tid) >> 5;           // grid exact: NT/8 blocks
    const int lane = tid & 31;
    const int o = vcol[t];
    const float2 hv = *(const float2*)(h + (size_t)o * EMB + 2 * lane);
    #pragma unroll
    for (int r = 0; r < RP; ++r) {
        const size_t e = (size_t)r * NT + t;
        const float lv = l2T[e];
        const int vr = vrow[e];
        const float wgt = lv / fmaxf(rowsum[vr], 1e-6f);
        const float cx = wgt * hv.x, cy = wgt * hv.y;
        if (vr == 0) {                                     // r=0 hot row
            atomicAdd(&acc0[2 * lane], cx);
            atomicAdd(&acc0[2 * lane + 1], cy);
        } else {
            atomicAdd(h2 + (size_t)vr * EMB + 2 * lane, cx);
            atomicAdd(h2 + (size_t)vr * EMB + 2 * lane + 1, cy);
        }
    }
    __syncthreads();
    if (tid < EMB) atomicAdd(h2 + tid, acc0[tid]);
}

// -------------------------------- logits = einsum('rhc,rnh->nc') via WMMA f32
// A = h2 viewed as (n, k=(r,h)) 16x512 tile ; B = weights2 flat (512x16) in LDS.
__global__ __launch_bounds__(256) void k_logits(
    const float* __restrict__ h2, const float* __restrict__ w2,
    const float* __restrict__ bias2, float* __restrict__ out)
{
    __shared__ float w2s[512 * 16];
    const int tid = threadIdx.x;
    for (int idx = tid; idx < 8192; idx += 256) w2s[idx] = w2[idx];
    __syncthreads();

    const int wave = tid >> 5, lane = tid & 31;
    const int m = lane & 15, g = lane >> 4;
    const int tile = blockIdx.x * 8 + wave;
    if (tile >= N_NODES / 16) return;                      // wave-uniform

    const int n0 = tile * 16;
    v8f c = {0.f, 0.f, 0.f, 0.f, 0.f, 0.f, 0.f, 0.f};
    #pragma unroll 4
    for (int k0 = 0; k0 < 512; k0 += 4) {
        const int k = k0 + 2 * g;                          // even; pair stays in row
        const int r = k >> 6, hh = k & 63;
        const float2 av =
            *(const float2*)(h2 + (size_t)(r * N_NODES + n0 + m) * EMB + hh);
        v2f a; a.x = av.x; a.y = av.y;
        v2f b; b.x = w2s[k * 16 + m]; b.y = w2s[(k + 1) * 16 + m];
        c = wmma4(a, b, c);
    }
    const float bv = bias2[m];
    #pragma unroll
    for (int v = 0; v < 8; ++v) {
        const int n = n0 + v + 8 * g;
        out[(size_t)n * N_CLASSES + m] = c[v] + bv;
    }
}

// ---------------------------------------------------------------------- launch
extern "C" void kernel_launch(void* const* d_in, const int* in_sizes, int n_in,
                              void* d_out, int out_size, void* d_ws, size_t ws_size,
                              hipStream_t stream)
{
    (void)in_sizes; (void)n_in; (void)out_size; (void)ws_size;
    const float* rm    = (const float*)d_in[0];
    const int*   hrow  = (const int*)d_in[1];
    const int*   hcol  = (const int*)d_in[2];
    const int*   vrow  = (const int*)d_in[3];
    const int*   vcol  = (const int*)d_in[4];
    const float* Wl1   = (const float*)d_in[5];
    const float* bl1   = (const float*)d_in[6];
    const float* Wl2   = (const float*)d_in[7];
    const float* bl2   = (const float*)d_in[8];
    const float* w1    = (const float*)d_in[9];
    const float* w2    = (const float*)d_in[10];
    const float* bias1 = (const float*)d_in[11];
    const float* bias2 = (const float*)d_in[12];
    float* out = (float*)d_out;

    float* ws     = (float*)d_ws;
    float* l1T    = ws;                                  // NE
    float* l2T    = l1T + (size_t)NE;                    // NE
    float* colsum = l2T + (size_t)NE;                    // NRP
    float* rowsum = colsum + (size_t)NRP;                // NRP
    float* h      = rowsum + (size_t)NRP;                // N_NODES*EMB
    float* h2     = h + (size_t)N_NODES * EMB;           // NRP*EMB

    const long zn4 = ((long)2 * NRP + (long)N_NODES * EMB + (long)NRP * EMB) / 4;
    k_zero<<<2048, 256, 0, stream>>>((float4*)colsum, zn4);
    k_l1l2<<<(NT / 16 + 7) / 8, 256, 0, stream>>>(rm, Wl1, bl1, Wl2, bl2, l1T, l2T);
    k_sums<<<2048, 256, 0, stream>>>(l1T, l2T, hcol, vrow, colsum, rowsum);
    k_hscatter<<<NT / 8, 256, 0, stream>>>(l1T, hrow, hcol, colsum, w1, h);
    k_hrelu<<<(N_NODES * EMB) / 256, 256, 0, stream>>>(h, bias1);
    k_h2scatter<<<NT / 8, 256, 0, stream>>>(l2T, vrow, vcol, rowsum, h, h2);
    k_logits<<<(N_NODES / 16 + 7) / 8, 256, 0, stream>>>(h2, w2, bias2, out);
}